// MCAN_44830868635747
// MI455X (gfx1250) — compile-verified
//
#include <hip/hip_runtime.h>
#include <hip/hip_bf16.h>
#include <math.h>

// ---------------- problem constants ----------------
constexpr int Bc  = 256;
constexpr int Lc  = 128;   // LQ == LD
constexpr int Dc  = 300;
constexpr int Hc  = 200;
constexpr int DIN = 312;   // D + 12
constexpr int G4  = 800;   // 4*H
#define NEGF (-3.402823466e38f)

// ---------------- WMMA fp32 16x16x4 ----------------
typedef __attribute__((ext_vector_type(2))) float v2f;
typedef __attribute__((ext_vector_type(8))) float v8f;

__device__ __forceinline__ v8f wmma_f32(v2f a, v2f b, v8f c) {
  // D = A(16x4) * B(4x16) + C(16x16), fp32 matrix pipe
  return __builtin_amdgcn_wmma_f32_16x16x4_f32(false, a, false, b, (short)0, c,
                                               false, false);
}

__device__ __forceinline__ float sigm(float x) { return 1.f / (1.f + expf(-x)); }

// ---------------------------------------------------------------------------
// Generic WMMA GEMM: C[b] = A[b](MxK) * B[b](KxN) (+bias[n]) (+relu)
// Wave computes a 16x64 strip: A fragment loaded once per K-step, reused for
// 4 WMMAs (4x fewer A loads, 4x fewer waves). Ragged N zero-filled per lane.
// A-frag: lane(l16,half): a = {A[m][k+2*half], A[m][k+2*half+1]}
// B-frag: lane(l16,half): b = {B[k+2*half][n], B[k+2*half+1][n]}
// C/D:    VGPR r, lane:   C[r + 8*half][n]
// M multiple of 16, K multiple of 4.
// ---------------------------------------------------------------------------
template<int TRANSB, int RELU>
__global__ void gemm_wmma(const float* __restrict__ A, int lda, long long sA,
                          const float* __restrict__ Bm, int ldb, long long sB,
                          float* __restrict__ C, int ldc, long long sC,
                          const float* __restrict__ bias,
                          int M, int N, int K)
{
  const int ntN = (N + 63) >> 6;                 // 64-wide N supertiles
  const int ntM = M >> 4;
  const int wave = threadIdx.x >> 5;
  const int tile = blockIdx.x * (blockDim.x >> 5) + wave;
  if (tile >= ntM * ntN) return;                 // wave-uniform
  const int mt = tile / ntN, nt = tile % ntN;
  const int lane = threadIdx.x & 31, half = lane >> 4, l16 = lane & 15;

  const long long bat = blockIdx.y;
  A  += bat * sA;  Bm += bat * sB;  C += bat * sC;

  const int m = mt * 16 + l16;
  const float* ap = A + (size_t)m * lda;

  int  n[4];
  bool nok[4];
  v8f  acc[4];
#pragma unroll
  for (int j = 0; j < 4; ++j) {
    n[j] = nt * 64 + j * 16 + l16;
    nok[j] = (n[j] < N);
    acc[j] = (v8f){0.f, 0.f, 0.f, 0.f, 0.f, 0.f, 0.f, 0.f};
  }

  for (int k0 = 0; k0 < K; k0 += 4) {
    const int ka = k0 + 2 * half;
    v2f a;  a.x = ap[ka];  a.y = ap[ka + 1];
    __builtin_prefetch(ap + ka + 32, 0, 0);      // global_prefetch_b8, stream A
#pragma unroll
    for (int j = 0; j < 4; ++j) {
      v2f bb;
      if (TRANSB) {
        bb.x = nok[j] ? Bm[(size_t)n[j] * ldb + ka]     : 0.f;
        bb.y = nok[j] ? Bm[(size_t)n[j] * ldb + ka + 1] : 0.f;
      } else {
        bb.x = nok[j] ? Bm[(size_t)ka * ldb + n[j]]       : 0.f;
        bb.y = nok[j] ? Bm[(size_t)(ka + 1) * ldb + n[j]] : 0.f;
      }
      acc[j] = wmma_f32(a, bb, acc[j]);
    }
  }

#pragma unroll
  for (int j = 0; j < 4; ++j) {
    if (!nok[j]) continue;
    const float bv = bias ? bias[n[j]] : 0.f;
#pragma unroll
    for (int r = 0; r < 8; ++r) {
      float v = acc[j][r] + bv;
      if (RELU) v = fmaxf(v, 0.f);
      C[(size_t)(mt * 16 + r + 8 * half) * ldc + n[j]] = v;
    }
  }
}

// ---------------------------------------------------------------------------
// Highway combine: out = sigmoid(t)*relu(h) + (1-sigmoid(t))*x
// ---------------------------------------------------------------------------
__global__ void mcan_highway(const float* __restrict__ x,
                             const float* __restrict__ hraw,
                             const float* __restrict__ traw,
                             float* __restrict__ out, long long n)
{
  long long i = (long long)blockIdx.x * blockDim.x + threadIdx.x;
  if (i >= n) return;
  float t = sigm(traw[i]);
  float h = fmaxf(hraw[i], 0.f);
  out[i] = t * h + (1.f - t) * x[i];
}

// ---------------------------------------------------------------------------
// 128-value shared-memory softmax helper (block of 128 threads)
// ---------------------------------------------------------------------------
__device__ __forceinline__ void softmax128(float* v, float* red)
{
  const int tid = threadIdx.x;
  red[tid] = v[tid]; __syncthreads();
  for (int s = 64; s > 0; s >>= 1) { if (tid < s) red[tid] = fmaxf(red[tid], red[tid + s]); __syncthreads(); }
  const float mx = red[0]; __syncthreads();
  const float e = expf(v[tid] - mx);
  red[tid] = e; __syncthreads();
  for (int s = 64; s > 0; s >>= 1) { if (tid < s) red[tid] += red[tid + s]; __syncthreads(); }
  const float inv = 1.f / red[0]; __syncthreads();
  v[tid] = e * inv; __syncthreads();
}

// ---------------------------------------------------------------------------
// Row softmax of masked S. grid=(L, B), block=128. In-place safe.
// mask: element (q,k) valid iff (!useRowMask || q<rlen[b]) && k<clen[b]
// ---------------------------------------------------------------------------
__global__ void mcan_softmax_rows(const float* __restrict__ Sin, float* __restrict__ Pout,
                                  const int* __restrict__ rlen, const int* __restrict__ clen,
                                  int useRowMask)
{
  const int b = blockIdx.y, q = blockIdx.x, tid = threadIdx.x;
  __shared__ float red[128];
  const int cl = clen[b];
  const bool rok = useRowMask ? (q < rlen[b]) : true;
  const long long off = ((long long)b * Lc + q) * Lc;
  float v = (rok && tid < cl) ? Sin[off + tid] : NEGF;
  red[tid] = v; __syncthreads();
  for (int s = 64; s > 0; s >>= 1) { if (tid < s) red[tid] = fmaxf(red[tid], red[tid + s]); __syncthreads(); }
  const float mx = red[0]; __syncthreads();
  const float e = expf(v - mx);
  red[tid] = e; __syncthreads();
  for (int s = 64; s > 0; s >>= 1) { if (tid < s) red[tid] += red[tid + s]; __syncthreads(); }
  Pout[off + tid] = e / red[0];
}

// Column softmax, written transposed: PT[b][k][q] = softmax_q(S_am[b][q][k])
__global__ void mcan_softmax_cols_T(const float* __restrict__ Sin, float* __restrict__ PT,
                                    const int* __restrict__ qlen, const int* __restrict__ dlen)
{
  const int b = blockIdx.y, k = blockIdx.x, tid = threadIdx.x;   // tid = q
  __shared__ float red[128];
  const int ql = qlen[b], dl = dlen[b];
  float v = (tid < ql && k < dl) ? Sin[((long long)b * Lc + tid) * Lc + k] : NEGF;
  red[tid] = v; __syncthreads();
  for (int s = 64; s > 0; s >>= 1) { if (tid < s) red[tid] = fmaxf(red[tid], red[tid + s]); __syncthreads(); }
  const float mx = red[0]; __syncthreads();
  const float e = expf(v - mx);
  red[tid] = e; __syncthreads();
  for (int s = 64; s > 0; s >>= 1) { if (tid < s) red[tid] += red[tid + s]; __syncthreads(); }
  PT[((long long)b * Lc + k) * Lc + tid] = e / red[0];
}

// ---------------------------------------------------------------------------
// Per-batch pooled vectors (max-pool / mean-pool attention). grid=B, block=128
// ---------------------------------------------------------------------------
__global__ void mcan_batch_stats(const float* __restrict__ S,
                                 const float* __restrict__ qo,
                                 const float* __restrict__ dox,
                                 const int* __restrict__ qlen, const int* __restrict__ dlen,
                                 float* __restrict__ qmaxp, float* __restrict__ qmeanp,
                                 float* __restrict__ dmaxp, float* __restrict__ dmeanp)
{
  const int b = blockIdx.x, tid = threadIdx.x;
  __shared__ float w[128];
  __shared__ float red[128];
  const int ql = qlen[b], dl = dlen[b];
  const float* Sb  = S   + (long long)b * Lc * Lc;
  const float* qob = qo  + (long long)b * Lc * Dc;
  const float* dob = dox + (long long)b * Lc * Dc;

  // ---- q max-pool ----
  { float mx = NEGF;
    if (tid < ql) for (int k = 0; k < dl; ++k) mx = fmaxf(mx, Sb[tid * Lc + k]);
    w[tid] = mx; }
  __syncthreads();
  softmax128(w, red);
  for (int d = tid; d < Dc; d += 128) {
    float a = 0.f;
    for (int q = 0; q < Lc; ++q) a += w[q] * qob[q * Dc + d];
    qmaxp[(long long)b * Dc + d] = a;
  }
  __syncthreads();
  // ---- q mean-pool ----
  { float s = 0.f;
    if (tid < ql) for (int k = 0; k < dl; ++k) s += Sb[tid * Lc + k];
    w[tid] = s / ((float)dl + 1e-15f); }
  __syncthreads();
  softmax128(w, red);
  for (int d = tid; d < Dc; d += 128) {
    float a = 0.f;
    for (int q = 0; q < Lc; ++q) a += w[q] * qob[q * Dc + d];
    qmeanp[(long long)b * Dc + d] = a;
  }
  __syncthreads();
  // ---- d max-pool ----
  { float mx = NEGF;
    if (tid < dl) for (int q = 0; q < ql; ++q) mx = fmaxf(mx, Sb[q * Lc + tid]);
    w[tid] = mx; }
  __syncthreads();
  softmax128(w, red);
  for (int d = tid; d < Dc; d += 128) {
    float a = 0.f;
    for (int k = 0; k < Lc; ++k) a += w[k] * dob[k * Dc + d];
    dmaxp[(long long)b * Dc + d] = a;
  }
  __syncthreads();
  // ---- d mean-pool ----
  { float s = 0.f;
    if (tid < dl) for (int q = 0; q < ql; ++q) s += Sb[q * Lc + tid];
    w[tid] = s / ((float)ql + 1e-15f); }
  __syncthreads();
  softmax128(w, red);
  for (int d = tid; d < Dc; d += 128) {
    float a = 0.f;
    for (int k = 0; k < Lc; ++k) a += w[k] * dob[k * Dc + d];
    dmeanp[(long long)b * Dc + d] = a;
  }
}

// ---------------------------------------------------------------------------
// Cast features + concat -> (B,L,312). grid=(L,B), block=128.
// out[:,:,0:12]  = 4 groups x [relu(cat(x,y)@Wc), relu((x*y)@Wm), relu((x-y)@Ws)]
// out[:,:,12:312]= xo (highway output)
// ---------------------------------------------------------------------------
__global__ void mcan_cast_concat(const float* __restrict__ emb, const float* __restrict__ xo,
                                 const float* __restrict__ maxp, const float* __restrict__ meanp,
                                 const float* __restrict__ align, const float* __restrict__ selfa,
                                 const float* __restrict__ Wc, const float* __restrict__ bc,
                                 const float* __restrict__ Wm, const float* __restrict__ bm,
                                 const float* __restrict__ Ws, const float* __restrict__ bs,
                                 int iBase, float* __restrict__ out)
{
  const int b = blockIdx.y, pos = blockIdx.x, tid = threadIdx.x;
  const long long po = (long long)b * Lc + pos;
  const float* y = emb + po * Dc;
  float* orow = out + po * DIN;
  __shared__ float r0[128], r1[128], r2[128];

  for (int g = 0; g < 4; ++g) {
    const float* Wci = Wc + (size_t)(iBase + g) * 2 * Dc;
    const float* Wmi = Wm + (size_t)(iBase + g) * Dc;
    const float* Wsi = Ws + (size_t)(iBase + g) * Dc;
    float s0 = 0.f, s1 = 0.f, s2 = 0.f;
    for (int d = tid; d < Dc; d += 128) {
      float xv;
      if (g == 0)      xv = maxp [(long long)b * Dc + d];
      else if (g == 1) xv = meanp[(long long)b * Dc + d];
      else if (g == 2) xv = align[po * Dc + d];
      else             xv = selfa[po * Dc + d];
      const float yv = y[d];
      s0 += xv * Wci[d] + yv * Wci[Dc + d];
      s1 += (xv * yv) * Wmi[d];
      s2 += (xv - yv) * Wsi[d];
    }
    r0[tid] = s0; r1[tid] = s1; r2[tid] = s2; __syncthreads();
    for (int s = 64; s > 0; s >>= 1) {
      if (tid < s) { r0[tid] += r0[tid + s]; r1[tid] += r1[tid + s]; r2[tid] += r2[tid + s]; }
      __syncthreads();
    }
    if (tid == 0) {
      orow[3 * g + 0] = fmaxf(r0[0] + bc[iBase + g], 0.f);
      orow[3 * g + 1] = fmaxf(r1[0] + bm[iBase + g], 0.f);
      orow[3 * g + 2] = fmaxf(r2[0] + bs[iBase + g], 0.f);
    }
    __syncthreads();
  }
  const float* xq = xo + po * Dc;
  for (int d = tid; d < Dc; d += 128) orow[12 + d] = xq[d];
}

// ---------------------------------------------------------------------------
// LSTM scan (persistent): 16 batch rows / block, 128 steps, recurrent GEMM
// (16x200)@(200x800) via in-block WMMA on LDS-staged h. grid=(B/16, 2 dirs).
// K-outer loop: each LDS A-fragment feeds the wave's 6-7 N-tiles (acc kept in
// VGPRs via full unroll + wave-uniform guards; EXEC stays all-ones for WMMA).
// hidden[b][t][dir*H + u]. xz already holds x@Wx + b.
// ---------------------------------------------------------------------------
__global__ void mcan_lstm_scan(const float* __restrict__ xz, long long xzStride,
                               const float* __restrict__ WhBase,
                               float* __restrict__ hidden, int L)
{
  __shared__ float hs[16 * Hc];    // 12.8 KB
  __shared__ float zs[16 * G4];    // 51.2 KB
  const int dir = blockIdx.y;
  const int b0  = blockIdx.x * 16;
  const float* Wh  = WhBase + (long long)dir * Hc * G4;
  const float* xzd = xz + (long long)dir * xzStride;
  const int tid = threadIdx.x;
  const int wave = tid >> 5, lane = tid & 31, half = lane >> 4, l16 = lane & 15;

  float creg[13];
#pragma unroll
  for (int j = 0; j < 13; ++j) creg[j] = 0.f;
  for (int i = tid; i < 16 * Hc; i += blockDim.x) hs[i] = 0.f;
  __syncthreads();

  for (int it = 0; it < L; ++it) {
    const int t = dir ? (L - 1 - it) : it;
    // z = h @ Wh : wave owns N-tiles nt = wave + 8*j (j < 7), 50 tiles total
    v8f acc[7];
#pragma unroll
    for (int j = 0; j < 7; ++j) acc[j] = (v8f){0.f, 0.f, 0.f, 0.f, 0.f, 0.f, 0.f, 0.f};
    for (int k0 = 0; k0 < Hc; k0 += 4) {
      const int ka = k0 + 2 * half;
      v2f a;  a.x = hs[l16 * Hc + ka];  a.y = hs[l16 * Hc + ka + 1];
#pragma unroll
      for (int j = 0; j < 7; ++j) {
        if (wave + 8 * j < G4 / 16) {                   // wave-uniform guard
          const int n = (wave + 8 * j) * 16 + l16;
          v2f bb; bb.x = Wh[(long long)ka * G4 + n];
                  bb.y = Wh[(long long)(ka + 1) * G4 + n];
          acc[j] = wmma_f32(a, bb, acc[j]);
        }
      }
    }
#pragma unroll
    for (int j = 0; j < 7; ++j) {
      if (wave + 8 * j < G4 / 16) {
        const int n = (wave + 8 * j) * 16 + l16;
#pragma unroll
        for (int r = 0; r < 8; ++r) zs[(r + 8 * half) * G4 + n] = acc[j][r];
      }
    }
    __syncthreads();
    // gates
    int j = 0;
    for (int e = tid; e < 16 * Hc; e += blockDim.x, ++j) {
      const int mb = e / Hc, u = e - mb * Hc;
      const long long ro = ((long long)(b0 + mb) * L + t) * G4;
      const float zi = xzd[ro + u]          + zs[mb * G4 + u];
      const float zf = xzd[ro + Hc + u]     + zs[mb * G4 + Hc + u];
      const float zg = xzd[ro + 2 * Hc + u] + zs[mb * G4 + 2 * Hc + u];
      const float zo = xzd[ro + 3 * Hc + u] + zs[mb * G4 + 3 * Hc + u];
      float c = creg[j];
      c = sigm(zf) * c + sigm(zi) * tanhf(zg);
      const float h = sigm(zo) * tanhf(c);
      creg[j] = c;
      hs[e] = h;
      hidden[((long long)(b0 + mb) * L + t) * (2 * Hc) + dir * Hc + u] = h;
    }
    __syncthreads();
  }
}

// ---------------------------------------------------------------------------
// Masked mean/max pooling over time. grid=B, block=256. fin[b] = [mean(400), max(400)]
// ---------------------------------------------------------------------------
__global__ void mcan_pool_final(const float* __restrict__ hid, const int* __restrict__ lens,
                                float* __restrict__ fin, int L)
{
  const int b = blockIdx.x, tid = threadIdx.x;
  const int len = lens[b];
  const float inv = 1.f / ((float)len + 1e-15f);
  for (int u = tid; u < 2 * Hc; u += blockDim.x) {
    float s = 0.f, mx = NEGF;
    const float* hp = hid + (long long)b * L * (2 * Hc) + u;
    for (int t = 0; t < L; ++t) {
      const float v = (t < len) ? hp[(long long)t * (2 * Hc)] : 0.f;
      s += v; mx = fmaxf(mx, v);
    }
    fin[(long long)b * 800 + u]       = s * inv;
    fin[(long long)b * 800 + 400 + u] = mx;
  }
}

// final = [qf, df, qf*df, qf-df]  (B x 3200). grid=B.
__global__ void mcan_build_final(const float* __restrict__ qf, const float* __restrict__ df,
                                 float* __restrict__ fin)
{
  const int b = blockIdx.x;
  float* o = fin + (long long)b * 3200;
  for (int u = threadIdx.x; u < 800; u += blockDim.x) {
    const float q = qf[(long long)b * 800 + u];
    const float d = df[(long long)b * 800 + u];
    o[u] = q; o[800 + u] = d; o[1600 + u] = q * d; o[2400 + u] = q - d;
  }
}

// ---------------------------------------------------------------------------
// Orchestration
// ---------------------------------------------------------------------------
extern "C" void kernel_launch(void* const* d_in, const int* in_sizes, int n_in,
                              void* d_out, int out_size, void* d_ws, size_t ws_size,
                              hipStream_t stream)
{
  const float* query_emb = (const float*)d_in[0];
  const float* doc_emb   = (const float*)d_in[1];
  const int*   qlen      = (const int*)d_in[2];
  const int*   dlen      = (const int*)d_in[3];
  const float* Mw        = (const float*)d_in[4];
  const float* hwq_Wt    = (const float*)d_in[5];
  const float* hwq_bt    = (const float*)d_in[6];
  const float* hwq_Wh    = (const float*)d_in[7];
  const float* hwq_bh    = (const float*)d_in[8];
  const float* hwd_Wt    = (const float*)d_in[9];
  const float* hwd_bt    = (const float*)d_in[10];
  const float* hwd_Wh    = (const float*)d_in[11];
  const float* hwd_bh    = (const float*)d_in[12];
  const float* cast_Wc   = (const float*)d_in[13];
  const float* cast_bc   = (const float*)d_in[14];
  const float* cast_Wm   = (const float*)d_in[15];
  const float* cast_bm   = (const float*)d_in[16];
  const float* cast_Ws   = (const float*)d_in[17];
  const float* cast_bs   = (const float*)d_in[18];
  const float* qWx       = (const float*)d_in[19];
  const float* qWh       = (const float*)d_in[20];
  const float* qb        = (const float*)d_in[21];
  const float* dWx       = (const float*)d_in[22];
  const float* dWh       = (const float*)d_in[23];
  const float* db        = (const float*)d_in[24];
  const float* fc1_W     = (const float*)d_in[25];
  const float* fc1_b     = (const float*)d_in[26];
  float* out = (float*)d_out;
  float* ws  = (float*)d_ws;

  // ---- workspace arena (floats) ----
  constexpr long long NE    = (long long)Bc * Lc * Dc;   //  9,830,400
  constexpr long long SS    = (long long)Bc * Lc * Lc;   //  4,194,304
  constexpr long long BD    = (long long)Bc * Dc;        //     76,800
  constexpr long long CASTN = (long long)Bc * Lc * DIN;  // 10,223,616
  constexpr long long XZSEG = (long long)Bc * Lc * G4;   // 26,214,400
  constexpr long long HIDN  = (long long)Bc * Lc * 2 * Hc;

  float* f_qo     = ws;
  float* f_do     = ws + 1 * NE;
  float* f_tA     = ws + 2 * NE;   // highway-h, then qoM
  float* f_tB     = ws + 3 * NE;   // highway-t
  float* f_qal    = ws + 4 * NE;
  float* f_dal    = ws + 5 * NE;
  float* f_qself  = ws + 6 * NE;
  float* f_dself  = ws + 7 * NE;
  float* f_S      = ws + 8 * NE;
  float* f_P      = f_S + SS;
  float* f_qmaxp  = f_P + SS;
  float* f_qmeanp = f_qmaxp + BD;
  float* f_dmaxp  = f_qmeanp + BD;
  float* f_dmeanp = f_dmaxp + BD;
  float* f_qcast  = f_dmeanp + BD;
  float* f_dcast  = f_qcast + CASTN;
  float* f_qhid   = f_dcast + CASTN;
  float* f_dhid   = f_qhid + HIDN;
  float* f_qfin   = f_dhid + HIDN;
  float* f_dfin   = f_qfin + (long long)Bc * 800;
  float* f_fin    = f_dfin + (long long)Bc * 800;
  float* f_xz     = ws;            // aliases attention region (dead by LSTM time)

  const int BL = Bc * Lc;          // 32768
  auto tiles = [](int M, int N) { return (M / 16) * ((N + 63) / 64); };

  // ---- 1. highway (q and d) ----
  {
    dim3 g((unsigned)((tiles(BL, Dc) + 3) / 4), 1, 1);
    gemm_wmma<0, 0><<<g, 128, 0, stream>>>(query_emb, Dc, 0, hwq_Wh, Dc, 0, f_tA, Dc, 0, hwq_bh, BL, Dc, Dc);
    gemm_wmma<0, 0><<<g, 128, 0, stream>>>(query_emb, Dc, 0, hwq_Wt, Dc, 0, f_tB, Dc, 0, hwq_bt, BL, Dc, Dc);
    mcan_highway<<<(unsigned)((NE + 255) / 256), 256, 0, stream>>>(query_emb, f_tA, f_tB, f_qo, NE);
    gemm_wmma<0, 0><<<g, 128, 0, stream>>>(doc_emb, Dc, 0, hwd_Wh, Dc, 0, f_tA, Dc, 0, hwd_bh, BL, Dc, Dc);
    gemm_wmma<0, 0><<<g, 128, 0, stream>>>(doc_emb, Dc, 0, hwd_Wt, Dc, 0, f_tB, Dc, 0, hwd_bt, BL, Dc, Dc);
    mcan_highway<<<(unsigned)((NE + 255) / 256), 256, 0, stream>>>(doc_emb, f_tA, f_tB, f_do, NE);
  }

  // ---- 2. qoM = qo @ M ----
  {
    dim3 g((unsigned)((tiles(BL, Dc) + 3) / 4), 1, 1);
    gemm_wmma<0, 0><<<g, 128, 0, stream>>>(f_qo, Dc, 0, Mw, Dc, 0, f_tA, Dc, 0, nullptr, BL, Dc, Dc);
  }

  // ---- 3. S[b] = qoM[b] @ do[b]^T ----
  {
    dim3 g((unsigned)((tiles(Lc, Lc) + 3) / 4), (unsigned)Bc, 1);
    gemm_wmma<1, 0><<<g, 128, 0, stream>>>(f_tA, Dc, (long long)Lc * Dc,
                                           f_do, Dc, (long long)Lc * Dc,
                                           f_S, Lc, (long long)Lc * Lc, nullptr, Lc, Lc, Dc);
  }

  // ---- 4. pooled max/mean attention vectors ----
  mcan_batch_stats<<<Bc, 128, 0, stream>>>(f_S, f_qo, f_do, qlen, dlen,
                                           f_qmaxp, f_qmeanp, f_dmaxp, f_dmeanp);

  // ---- 5. q_align = softmax_k(S_am) @ do ----
  mcan_softmax_rows<<<dim3(Lc, Bc), 128, 0, stream>>>(f_S, f_P, qlen, dlen, 1);
  {
    dim3 g((unsigned)((tiles(Lc, Dc) + 3) / 4), (unsigned)Bc, 1);
    gemm_wmma<0, 0><<<g, 128, 0, stream>>>(f_P, Lc, (long long)Lc * Lc,
                                           f_do, Dc, (long long)Lc * Dc,
                                           f_qal, Dc, (long long)Lc * Dc, nullptr, Lc, Dc, Lc);
  }

  // ---- 6. d_align = softmax_q(S_am)^T @ qo ----
  mcan_softmax_cols_T<<<dim3(Lc, Bc), 128, 0, stream>>>(f_S, f_P, qlen, dlen);
  {
    dim3 g((unsigned)((tiles(Lc, Dc) + 3) / 4), (unsigned)Bc, 1);
    gemm_wmma<0, 0><<<g, 128, 0, stream>>>(f_P, Lc, (long long)Lc * Lc,
                                           f_qo, Dc, (long long)Lc * Dc,
                                           f_dal, Dc, (long long)Lc * Dc, nullptr, Lc, Dc, Lc);
  }

  // ---- 7. self-attention (q then d), key-mask only ----
  {
    dim3 gs((unsigned)((tiles(Lc, Lc) + 3) / 4), (unsigned)Bc, 1);
    dim3 ga((unsigned)((tiles(Lc, Dc) + 3) / 4), (unsigned)Bc, 1);
    gemm_wmma<1, 0><<<gs, 128, 0, stream>>>(f_qo, Dc, (long long)Lc * Dc,
                                            f_qo, Dc, (long long)Lc * Dc,
                                            f_P, Lc, (long long)Lc * Lc, nullptr, Lc, Lc, Dc);
    mcan_softmax_rows<<<dim3(Lc, Bc), 128, 0, stream>>>(f_P, f_P, qlen, qlen, 0);
    gemm_wmma<0, 0><<<ga, 128, 0, stream>>>(f_P, Lc, (long long)Lc * Lc,
                                            f_qo, Dc, (long long)Lc * Dc,
                                            f_qself, Dc, (long long)Lc * Dc, nullptr, Lc, Dc, Lc);
    gemm_wmma<1, 0><<<gs, 128, 0, stream>>>(f_do, Dc, (long long)Lc * Dc,
                                            f_do, Dc, (long long)Lc * Dc,
                                            f_P, Lc, (long long)Lc * Lc, nullptr, Lc, Lc, Dc);
    mcan_softmax_rows<<<dim3(Lc, Bc), 128, 0, stream>>>(f_P, f_P, dlen, dlen, 0);
    gemm_wmma<0, 0><<<ga, 128, 0, stream>>>(f_P, Lc, (long long)Lc * Lc,
                                            f_do, Dc, (long long)Lc * Dc,
                                            f_dself, Dc, (long long)Lc * Dc, nullptr, Lc, Dc, Lc);
  }

  // ---- 8. cast features + concat ----
  mcan_cast_concat<<<dim3(Lc, Bc), 128, 0, stream>>>(query_emb, f_qo, f_qmaxp, f_qmeanp, f_qal, f_qself,
                                                     cast_Wc, cast_bc, cast_Wm, cast_bm, cast_Ws, cast_bs,
                                                     0, f_qcast);
  mcan_cast_concat<<<dim3(Lc, Bc), 128, 0, stream>>>(doc_emb, f_do, f_dmaxp, f_dmeanp, f_dal, f_dself,
                                                     cast_Wc, cast_bc, cast_Wm, cast_bm, cast_Ws, cast_bs,
                                                     4, f_dcast);

  // ---- 9. q LSTM (xz GEMMs then scan); xz aliases dead attention buffers ----
  {
    dim3 g((unsigned)((tiles(BL, G4) + 3) / 4), 1, 1);
    gemm_wmma<0, 0><<<g, 128, 0, stream>>>(f_qcast, DIN, 0, qWx, G4, 0,
                                           f_xz, G4, 0, qb, BL, G4, DIN);
    gemm_wmma<0, 0><<<g, 128, 0, stream>>>(f_qcast, DIN, 0, qWx + (long long)DIN * G4, G4, 0,
                                           f_xz + XZSEG, G4, 0, qb + G4, BL, G4, DIN);
    mcan_lstm_scan<<<dim3(Bc / 16, 2), 256, 0, stream>>>(f_xz, XZSEG, qWh, f_qhid, Lc);
  }
  // ---- 10. d LSTM ----
  {
    dim3 g((unsigned)((tiles(BL, G4) + 3) / 4), 1, 1);
    gemm_wmma<0, 0><<<g, 128, 0, stream>>>(f_dcast, DIN, 0, dWx, G4, 0,
                                           f_xz, G4, 0, db, BL, G4, DIN);
    gemm_wmma<0, 0><<<g, 128, 0, stream>>>(f_dcast, DIN, 0, dWx + (long long)DIN * G4, G4, 0,
                                           f_xz + XZSEG, G4, 0, db + G4, BL, G4, DIN);
    mcan_lstm_scan<<<dim3(Bc / 16, 2), 256, 0, stream>>>(f_xz, XZSEG, dWh, f_dhid, Lc);
  }

  // ---- 11. pooling + final feature vector ----
  mcan_pool_final<<<Bc, 256, 0, stream>>>(f_qhid, qlen, f_qfin, Lc);
  mcan_pool_final<<<Bc, 256, 0, stream>>>(f_dhid, dlen, f_dfin, Lc);
  mcan_build_final<<<Bc, 256, 0, stream>>>(f_qfin, f_dfin, f_fin);

  // ---- 12. yout = relu(final @ fc1_W + fc1_b) ----
  {
    dim3 g((unsigned)((tiles(Bc, Hc) + 3) / 4), 1, 1);
    gemm_wmma<0, 1><<<g, 128, 0, stream>>>(f_fin, 16 * Hc, 0, fc1_W, Hc, 0,
                                           out, Hc, 0, fc1_b, Bc, Hc, 16 * Hc);
  }
  (void)in_sizes; (void)n_in; (void)out_size; (void)ws_size;
}